// GCN_2_Paper_3246995276083
// MI455X (gfx1250) — compile-verified
//
#include <hip/hip_runtime.h>

typedef __attribute__((ext_vector_type(2))) float v2f;
typedef __attribute__((ext_vector_type(8))) float v8f;

#define INDIM 512
#define HID   16
#define OUTD  64

// D = A(16x4) * B(4x16) + C, f32, wave32 WMMA
static __device__ __forceinline__ v8f wmma_f32_k4(v2f a, v2f b, v8f c) {
  return __builtin_amdgcn_wmma_f32_16x16x4_f32(false, a, false, b, (short)0, c,
                                               false, false);
}

// ---------------- utility kernels ----------------
__global__ void zero_f32(float* __restrict__ p, long n) {
  long i = (long)blockIdx.x * blockDim.x + threadIdx.x;
  if (i < n) p[i] = 0.0f;
}

__global__ void deg_count(const int* __restrict__ dstE, float* __restrict__ deg,
                          int nE) {
  int e = blockIdx.x * blockDim.x + threadIdx.x;
  if (e < nE) atomicAdd(&deg[dstE[e]], 1.0f);  // exact: deg < 2^24
}

__global__ void make_dinv(float* __restrict__ d, int n) {
  int v = blockIdx.x * blockDim.x + threadIdx.x;
  if (v < n) d[v] = rsqrtf(d[v] + 1.0f);  // +1 self-loop; always > 0
}

// ---------------- GEMM1: H1 = X @ W1  (Nx512 * 512x16) ----------------
// one wave per 16-row tile; W1 staged in LDS; 128 K-steps of 16x16x4 WMMA
__global__ void gemm1_wmma(const float* __restrict__ X,
                           const float* __restrict__ W1,
                           float* __restrict__ H1, int n_tiles) {
  __shared__ float w[INDIM * HID];  // 32 KB
  for (int i = threadIdx.x; i < INDIM * HID / 4; i += blockDim.x)
    ((float4*)w)[i] = ((const float4*)W1)[i];
  __syncthreads();

  int wave = (blockIdx.x * blockDim.x + threadIdx.x) >> 5;
  int lane = threadIdx.x & 31;
  if (wave >= n_tiles) return;  // wave-uniform exit
  int half = lane >> 4;         // 0: K,K+1   1: K+2,K+3
  int col  = lane & 15;         // A row within tile == B column
  long row = (long)wave * 16 + col;
  const float* xr = X + row * INDIM;

  v8f c = {};
  for (int kb = 0; kb < INDIM; kb += 4) {
    v2f a = *(const v2f*)(xr + kb + 2 * half);
    v2f b;
    b.x = w[(kb + 2 * half) * HID + col];
    b.y = w[(kb + 2 * half + 1) * HID + col];
    c = wmma_f32_k4(a, b, c);
  }
  long base = (long)wave * 16;
#pragma unroll
  for (int r = 0; r < 8; ++r) {
    int m = half ? (r + 8) : r;   // C layout: VGPR r -> M=r / M=r+8
    H1[(base + m) * HID + col] = c[r];
  }
}

__global__ void gemm1_tail(const float* __restrict__ X,
                           const float* __restrict__ W1,
                           float* __restrict__ H1, int row0, int n) {
  int t = blockIdx.x * blockDim.x + threadIdx.x;
  int nrows = n - row0;
  if (t >= nrows * HID) return;
  int r = row0 + t / HID, c = t % HID;
  float acc = 0.0f;
  for (int k = 0; k < INDIM; ++k) acc += X[(long)r * INDIM + k] * W1[k * HID + c];
  H1[(long)r * HID + c] = acc;
}

// ---------------- edge scatter: agg[dst] += h[src]*dinv[s]*dinv[d] ----------
// lane = (edge, channel); half-wave does 64B coalesced gather + grouped atomics
__global__ void edge_scatter(const int* __restrict__ srcE,
                             const int* __restrict__ dstE,
                             const float* __restrict__ h,
                             const float* __restrict__ dinv,
                             float* __restrict__ agg, long total) {
  long t = (long)blockIdx.x * blockDim.x + threadIdx.x;
  if (t >= total) return;
  long e = t >> 4;
  int  c = (int)(t & 15);
  int s = srcE[e], d = dstE[e];
  float nrm = dinv[s] * dinv[d];
  atomicAdd(&agg[(long)d * HID + c], h[(long)s * HID + c] * nrm);
}

// ---------------- H = relu(agg1 + H1*dinv^2 + b1)   (in place on agg1) ------
__global__ void relu_bias_selfloop(float* __restrict__ agg1,
                                   const float* __restrict__ H1,
                                   const float* __restrict__ dinv,
                                   const float* __restrict__ b1, long total) {
  long t = (long)blockIdx.x * blockDim.x + threadIdx.x;
  if (t >= total) return;
  long i = t >> 4;
  int  c = (int)(t & 15);
  float dv = dinv[i];
  float v = agg1[t] + H1[t] * dv * dv + b1[c];
  agg1[t] = fmaxf(v, 0.0f);
}

// ---------------- GEMM2: out = (agg2 + H*dinv^2) @ W2 + b2  (Nx16 * 16x64) --
__global__ void gemm2_wmma(const float* __restrict__ agg2,
                           const float* __restrict__ H,
                           const float* __restrict__ dinv,
                           const float* __restrict__ W2,
                           const float* __restrict__ b2,
                           float* __restrict__ out, int n_tiles) {
  int wave = (blockIdx.x * blockDim.x + threadIdx.x) >> 5;
  int lane = threadIdx.x & 31;
  if (wave >= n_tiles) return;
  int half = lane >> 4;
  int col  = lane & 15;
  long R = (long)wave * 16 + col;
  float dv  = dinv[R];
  float dv2 = dv * dv;

  v2f a[4];  // fused self-loop: A row = agg2 + H*dinv^2
#pragma unroll
  for (int j = 0; j < 4; ++j) {
    int k = 4 * j + 2 * half;
    v2f g = *(const v2f*)(agg2 + R * HID + k);
    v2f h = *(const v2f*)(H + R * HID + k);
    a[j].x = g.x + h.x * dv2;
    a[j].y = g.y + h.y * dv2;
  }

  long base = (long)wave * 16;
#pragma unroll
  for (int nt = 0; nt < 4; ++nt) {  // four 16-col output tiles
    v8f c = {};
#pragma unroll
    for (int j = 0; j < 4; ++j) {
      int k = 4 * j + 2 * half;
      v2f b;
      b.x = W2[k * OUTD + nt * 16 + col];        // W2: 4KB, cache-resident
      b.y = W2[(k + 1) * OUTD + nt * 16 + col];
      c = wmma_f32_k4(a[j], b, c);
    }
    float bias = b2[nt * 16 + col];
#pragma unroll
    for (int r = 0; r < 8; ++r) {
      int m = half ? (r + 8) : r;
      out[(base + m) * OUTD + nt * 16 + col] = c[r] + bias;
    }
  }
}

__global__ void gemm2_tail(const float* __restrict__ agg2,
                           const float* __restrict__ H,
                           const float* __restrict__ dinv,
                           const float* __restrict__ W2,
                           const float* __restrict__ b2,
                           float* __restrict__ out, int row0, int n) {
  int t = blockIdx.x * blockDim.x + threadIdx.x;
  int nrows = n - row0;
  if (t >= nrows * OUTD) return;
  int r = row0 + t / OUTD, c = t % OUTD;
  float dv = dinv[r], dv2 = dv * dv;
  float acc = b2[c];
  for (int k = 0; k < HID; ++k)
    acc += (agg2[(long)r * HID + k] + H[(long)r * HID + k] * dv2) * W2[k * OUTD + c];
  out[(long)r * OUTD + c] = acc;
}

// ---------------- driver ----------------
extern "C" void kernel_launch(void* const* d_in, const int* in_sizes, int n_in,
                              void* d_out, int out_size, void* d_ws,
                              size_t ws_size, hipStream_t stream) {
  (void)n_in; (void)out_size; (void)ws_size;
  const int*   E  = (const int*)d_in[1];
  const float* X  = (const float*)d_in[2];
  const float* W1 = (const float*)d_in[3];
  const float* b1 = (const float*)d_in[4];
  const float* W2 = (const float*)d_in[5];
  const float* b2 = (const float*)d_in[6];
  float* out = (float*)d_out;

  const int n  = in_sizes[0];
  const int nE = in_sizes[1] / 2;
  const int* srcE = E;
  const int* dstE = E + nE;

  // workspace layout (floats): [dinv n][agg1 16n][agg2 16n][H1 16n]
  float* w    = (float*)d_ws;
  float* dinv = w;
  float* agg1 = w + n;                 // becomes H after relu (in place)
  float* agg2 = agg1 + (long)n * HID;
  float* H1   = agg2 + (long)n * HID;

  const int B = 256;

  // 0) zero dinv+agg1+agg2 (contiguous 33n floats) every call
  long zn = (long)n * (1 + 2 * HID);
  zero_f32<<<(unsigned)((zn + B - 1) / B), B, 0, stream>>>(w, zn);

  // 1) degrees -> dinv
  deg_count<<<(nE + B - 1) / B, B, 0, stream>>>(dstE, dinv, nE);
  make_dinv<<<(n + B - 1) / B, B, 0, stream>>>(dinv, n);

  // 2) H1 = X @ W1 (WMMA f32 16x16x4)
  int n_tiles = n / 16;
  int waves_per_blk = B / 32;
  if (n_tiles > 0)
    gemm1_wmma<<<(n_tiles + waves_per_blk - 1) / waves_per_blk, B, 0, stream>>>(
        X, W1, H1, n_tiles);
  int rem_rows = n - n_tiles * 16;
  if (rem_rows > 0)
    gemm1_tail<<<(rem_rows * HID + B - 1) / B, B, 0, stream>>>(X, W1, H1,
                                                               n_tiles * 16, n);

  // 3) layer-1 aggregation (L2-resident gather/scatter)
  long tot = (long)nE * HID;
  edge_scatter<<<(unsigned)((tot + B - 1) / B), B, 0, stream>>>(srcE, dstE, H1,
                                                                dinv, agg1, tot);

  // 4) H = relu(agg1 + selfloop + b1), in place
  long totN = (long)n * HID;
  relu_bias_selfloop<<<(unsigned)((totN + B - 1) / B), B, 0, stream>>>(
      agg1, H1, dinv, b1, totN);

  // 5) layer-2 aggregation on 16-wide H (projection reassociated to after agg)
  edge_scatter<<<(unsigned)((tot + B - 1) / B), B, 0, stream>>>(srcE, dstE, agg1,
                                                                dinv, agg2, tot);

  // 6) out = (agg2 + H*dinv^2) @ W2 + b2 (WMMA f32)
  if (n_tiles > 0)
    gemm2_wmma<<<(n_tiles + waves_per_blk - 1) / waves_per_blk, B, 0, stream>>>(
        agg2, agg1, dinv, W2, b2, out, n_tiles);
  if (rem_rows > 0)
    gemm2_tail<<<(rem_rows * OUTD + B - 1) / B, B, 0, stream>>>(
        agg2, agg1, dinv, W2, b2, out, n_tiles * 16, n);
}